// Scorer_8847632630252
// MI455X (gfx1250) — compile-verified
//
#include <hip/hip_runtime.h>
#include <hip/hip_bf16.h>

#define S_DIM 512
#define B_DIM 128
#define E_DIM 1024
#define H4_DIM 2048
#define OUT_DIM 1024

typedef __attribute__((ext_vector_type(2))) float v2f;
typedef __attribute__((ext_vector_type(8))) float v8f;

// ---------------------------------------------------------------------------
// Kernel 1: per-batch fused attention.
// One workgroup (256 threads = 8 waves) per batch b.
// Pass 1: scores[s] = dot(enc[s,b,:], w_enc) + dot(state[b,:], w_state) + bias
// Softmax over S in LDS (deterministic tree reductions).
// Pass 2: attended[b,e] = sum_s w[s] * enc[s,b,e]  (L2-resident re-read)
// Writes full_input[b, 0:1024]=attended, full_input[b, 1024:2048]=state.
// ---------------------------------------------------------------------------
__global__ __launch_bounds__(256) void attn_fused_kernel(
    const float* __restrict__ enc,     // (S,B,E)
    const float* __restrict__ state,   // (B,E)
    const float* __restrict__ attn_w,  // (H4,)
    const float* __restrict__ attn_b,  // (1,)
    float* __restrict__ full_input)    // (B, 2E)
{
    __shared__ float s_wenc[E_DIM];     // 4 KB
    __shared__ float s_scores[S_DIM];   // 2 KB
    __shared__ float s_red[256];
    __shared__ float s_bcast;

    const int b    = blockIdx.x;
    const int t    = threadIdx.x;
    const int lane = t & 31;
    const int wave = t >> 5;

    // preload w_enc into LDS
    for (int i = t; i < E_DIM; i += 256) s_wenc[i] = attn_w[E_DIM + i];

    // state_score = dot(state[b,:], w_state), tree-reduced (deterministic)
    const float* srow = state + (size_t)b * E_DIM;
    float ss = 0.f;
    for (int i = t; i < E_DIM; i += 256) ss += srow[i] * attn_w[i];
    s_red[t] = ss;
    __syncthreads();
    for (int off = 128; off > 0; off >>= 1) {
        if (t < off) s_red[t] += s_red[t + off];
        __syncthreads();
    }
    if (t == 0) s_bcast = s_red[0] + attn_b[0];
    __syncthreads();
    const float state_score = s_bcast;
    __syncthreads();

    // Pass 1: one wave per row s, strided over 8 waves; float4 loads.
    for (int s = wave; s < S_DIM; s += 8) {
        const float4* row = (const float4*)(enc + ((size_t)s * B_DIM + b) * E_DIM);
        const float4* w4  = (const float4*)s_wenc;
        float acc = 0.f;
#pragma unroll
        for (int j = 0; j < (E_DIM / 4) / 32; ++j) {   // 8 iterations
            float4 v = row[lane + 32 * j];
            float4 w = w4[lane + 32 * j];
            acc += v.x * w.x + v.y * w.y + v.z * w.z + v.w * w.w;
        }
        for (int off = 16; off > 0; off >>= 1)
            acc += __shfl_down(acc, off, 32);
        if (lane == 0) s_scores[s] = acc + state_score;
    }
    __syncthreads();

    // Softmax over 512 scores: thread t owns s=t and s=t+256.
    float a0 = s_scores[t], a1 = s_scores[t + 256];
    s_red[t] = fmaxf(a0, a1);
    __syncthreads();
    for (int off = 128; off > 0; off >>= 1) {
        if (t < off) s_red[t] = fmaxf(s_red[t], s_red[t + off]);
        __syncthreads();
    }
    const float gmax = s_red[0];
    __syncthreads();
    const float e0 = __expf(a0 - gmax);
    const float e1 = __expf(a1 - gmax);
    s_red[t] = e0 + e1;
    __syncthreads();
    for (int off = 128; off > 0; off >>= 1) {
        if (t < off) s_red[t] += s_red[t + off];
        __syncthreads();
    }
    const float inv = 1.0f / s_red[0];
    __syncthreads();
    s_scores[t]       = e0 * inv;
    s_scores[t + 256] = e1 * inv;
    __syncthreads();

    // Pass 2: thread t owns e = 4t..4t+3; stream all S rows (L2 hits).
    const float4* p = (const float4*)enc + (size_t)b * (E_DIM / 4) + t;
    const size_t  stride = (size_t)B_DIM * (E_DIM / 4);
    float4 acc = make_float4(0.f, 0.f, 0.f, 0.f);
#pragma unroll 4
    for (int s = 0; s < S_DIM; ++s) {
        const float  w = s_scores[s];
        const float4 v = p[(size_t)s * stride];
        acc.x += w * v.x; acc.y += w * v.y; acc.z += w * v.z; acc.w += w * v.w;
    }
    float* out_row = full_input + (size_t)b * H4_DIM;
    ((float4*)out_row)[t]                 = acc;                       // attended
    ((float4*)(out_row + E_DIM))[t]       = ((const float4*)srow)[t];  // state
}

// ---------------------------------------------------------------------------
// Kernel 2: C(128x1024) = full_input(128x2048) @ scorer_w(2048x1024) + bias
// Exact fp32 via V_WMMA_F32_16X16X4_F32. One wave per 16x16 tile:
// 8 M-tiles x 64 N-tiles = 512 waves = 64 blocks of 8 waves.
// Fragment layouts per ISA 7.12.2:
//   A (16x4):  lane l, vgpr v -> A[l&15][k + v + 2*(l>>4)]
//   B (4x16):  lane l, vgpr v -> B[k + v + 2*(l>>4)][l&15]
//   C (16x16): lane l, vgpr v -> C[v + 8*(l>>4)][l&15]
// Bias folded into accumulator init (depends only on N = l&15).
// ---------------------------------------------------------------------------
__global__ __launch_bounds__(256) void scorer_gemm_kernel(
    const float* __restrict__ A,     // (128, 2048)
    const float* __restrict__ Bm,    // (2048, 1024)
    const float* __restrict__ bias,  // (1024,)
    float* __restrict__ C)           // (128, 1024)
{
    const int t    = threadIdx.x;
    const int lane = t & 31;
    const int wave = t >> 5;
    const int tile = blockIdx.x * 8 + wave;   // 0..511
    const int m0   = (tile & 7)  * 16;
    const int n0   = (tile >> 3) * 16;

    const int lo = lane & 15;
    const int hi = lane >> 4;     // 0 or 1

    v8f acc;
    const float bv = bias[n0 + lo];
#pragma unroll
    for (int i = 0; i < 8; ++i) acc[i] = bv;

    const float* arow = A  + (size_t)(m0 + lo) * H4_DIM;
    const float* bcol = Bm + n0 + lo;

#pragma unroll 4
    for (int k = 0; k < H4_DIM; k += 4) {
        const int ka = k + 2 * hi;
        const float2 a2 = *(const float2*)(arow + ka);   // K=ka, ka+1
        v2f a; a[0] = a2.x; a[1] = a2.y;
        v2f bf;
        bf[0] = bcol[(size_t)ka       * OUT_DIM];
        bf[1] = bcol[(size_t)(ka + 1) * OUT_DIM];
        acc = __builtin_amdgcn_wmma_f32_16x16x4_f32(
            /*neg_a=*/false, a, /*neg_b=*/false, bf,
            /*c_mod=*/(short)0, acc, /*reuse_a=*/false, /*reuse_b=*/false);
    }

    float* crow = C + (size_t)(m0 + 8 * hi) * OUT_DIM + n0 + lo;
#pragma unroll
    for (int v = 0; v < 8; ++v) crow[(size_t)v * OUT_DIM] = acc[v];
}

extern "C" void kernel_launch(void* const* d_in, const int* in_sizes, int n_in,
                              void* d_out, int out_size, void* d_ws, size_t ws_size,
                              hipStream_t stream) {
    const float* enc      = (const float*)d_in[0];
    const float* state    = (const float*)d_in[1];
    const float* attn_w   = (const float*)d_in[2];
    const float* attn_b   = (const float*)d_in[3];
    const float* scorer_w = (const float*)d_in[4];
    const float* scorer_b = (const float*)d_in[5];
    float* out        = (float*)d_out;
    float* full_input = (float*)d_ws;   // (B, 2048) = 1 MB scratch

    attn_fused_kernel<<<B_DIM, 256, 0, stream>>>(enc, state, attn_w, attn_b, full_input);
    scorer_gemm_kernel<<<64, 256, 0, stream>>>(full_input, scorer_w, scorer_b, out);
}